// MultiLayerRNNModel_61452392071468
// MI455X (gfx1250) — compile-verified
//
#include <hip/hip_runtime.h>

// 2-layer tanh RNN, T=2048, B=4096, I=3, H=5.
// Roofline: ~268 MB mandatory traffic (read x 100MB, write out 168MB) at 23.3 TB/s
// => ~11.5us floor; FLOPs (~1.5G) are negligible. Strategy: fully fused recurrence,
// one wave per 16-batch tile (256 waves), transposed state kept in WMMA B-layout,
// all GEMMs via V_WMMA_F32_16X16X4_F32 chained accumulation, bias preloaded in C.
// tanh via v_exp_f32 + v_rcp_f32 (no IEEE-div expansion on the serial critical path).

typedef float v2f __attribute__((ext_vector_type(2)));
typedef float v8f __attribute__((ext_vector_type(8)));

#define T_LEN 2048
#define BATCH 4096
#define ISZ   3
#define HSZ   5

__device__ __forceinline__ float fast_tanh(float x) {
  float ax = __builtin_fabsf(x);
  float e  = __expf(-2.0f * ax);                       // v_exp_f32
  float r  = (1.0f - e) * __builtin_amdgcn_rcpf(1.0f + e); // v_rcp_f32, no div expansion
  return __builtin_copysignf(r, x);
}

// broadcast value from lane (lane & 15)  (pull cols 0..15 into the upper half-wave)
__device__ __forceinline__ float shfl16(float v, int l16) {
  return __shfl(v, l16, 32);
}

__global__ void __launch_bounds__(32)
rnn2_wmma_kernel(const float* __restrict__ x,
                 const float* __restrict__ hx,
                 const float* __restrict__ w_ih0, const float* __restrict__ w_hh0,
                 const float* __restrict__ b_ih0, const float* __restrict__ b_hh0,
                 const float* __restrict__ w_ih1, const float* __restrict__ w_hh1,
                 const float* __restrict__ b_ih1, const float* __restrict__ b_hh1,
                 float* __restrict__ out)
{
  const int lane = threadIdx.x & 31;
  const int half = lane >> 4;      // 0 = lanes 0-15, 1 = lanes 16-31
  const int l16  = lane & 15;
  const int b0   = blockIdx.x * 16; // batch tile base (16 columns per wave)

  // ---------------- constant A operands (weights) in WMMA A-layout -------------
  // A (16x4 f32): lanes 0-15: M=lane, v.x=K(k0+0), v.y=K(k0+1)
  //               lanes16-31: M=lane-16, v.x=K(k0+2), v.y=K(k0+3)
  auto ldw = [](const float* w, int rows, int cols, int r, int c) -> float {
    return (r < rows && c < cols) ? w[r * cols + c] : 0.0f;
  };
  const int am = l16;
  const int kb = half * 2;

  v2f a_ih0, a_hh0_0, a_hh0_1, a_ih1_0, a_ih1_1, a_hh1_0, a_hh1_1;
  a_ih0.x   = ldw(w_ih0, HSZ, ISZ, am, kb + 0);
  a_ih0.y   = ldw(w_ih0, HSZ, ISZ, am, kb + 1);
  a_hh0_0.x = ldw(w_hh0, HSZ, HSZ, am, kb + 0);
  a_hh0_0.y = ldw(w_hh0, HSZ, HSZ, am, kb + 1);
  a_hh0_1.x = ldw(w_hh0, HSZ, HSZ, am, 4 + kb + 0);
  a_hh0_1.y = ldw(w_hh0, HSZ, HSZ, am, 4 + kb + 1);
  a_ih1_0.x = ldw(w_ih1, HSZ, HSZ, am, kb + 0);
  a_ih1_0.y = ldw(w_ih1, HSZ, HSZ, am, kb + 1);
  a_ih1_1.x = ldw(w_ih1, HSZ, HSZ, am, 4 + kb + 0);
  a_ih1_1.y = ldw(w_ih1, HSZ, HSZ, am, 4 + kb + 1);
  a_hh1_0.x = ldw(w_hh1, HSZ, HSZ, am, kb + 0);
  a_hh1_0.y = ldw(w_hh1, HSZ, HSZ, am, kb + 1);
  a_hh1_1.x = ldw(w_hh1, HSZ, HSZ, am, 4 + kb + 0);
  a_hh1_1.y = ldw(w_hh1, HSZ, HSZ, am, 4 + kb + 1);

  // ---------------- bias C operands (row m gets bias[m], only lanes 0-15) ------
  v8f cb0, cb1;
  #pragma unroll
  for (int j = 0; j < 8; ++j) {
    float v0 = (j < HSZ) ? (b_ih0[j] + b_hh0[j]) : 0.0f;
    float v1 = (j < HSZ) ? (b_ih1[j] + b_hh1[j]) : 0.0f;
    cb0[j] = half ? 0.0f : v0;   // lanes 16-31 are rows 8-15 -> zero
    cb1[j] = half ? 0.0f : v1;
  }

  // ---------------- initial states (transposed) in WMMA B-layout ---------------
  // B (4x16 f32): v.x = row (k0+0) [lanes 0-15] / row (k0+2) [lanes 16-31]
  //               v.y = row (k0+1) [lanes 0-15] / row (k0+3) [lanes 16-31]
  const size_t h0b = (size_t)(b0 + l16) * HSZ;           // hx[0][b][.]
  const size_t h1b = ((size_t)BATCH + b0 + l16) * HSZ;   // hx[1][b][.]
  v2f s0c0, s0c1, s1c0, s1c1;
  s0c0.x = hx[h0b + kb + 0];
  s0c0.y = hx[h0b + kb + 1];
  { float t4 = hx[h0b + 4]; s0c1.x = half ? 0.0f : t4; }  // K=4 (lo) / K=6 (pad)
  s0c1.y = 0.0f;                                          // K=5 / K=7 pad
  s1c0.x = hx[h1b + kb + 0];
  s1c0.y = hx[h1b + kb + 1];
  { float t4 = hx[h1b + 4]; s1c1.x = half ? 0.0f : t4; }
  s1c1.y = 0.0f;

  float e0 = 0.f, e1 = 0.f, e2 = 0.f, e3 = 0.f, e4 = 0.f; // last layer-0 state
  float f0 = 0.f, f1 = 0.f, f2 = 0.f, f3 = 0.f, f4 = 0.f; // last layer-1 state

  for (int t = 0; t < T_LEN; ++t) {
    const size_t xb = ((size_t)t * BATCH + b0 + l16) * ISZ;
    if (t + 6 < T_LEN)
      __builtin_prefetch(x + xb + (size_t)6 * BATCH * ISZ, 0, 0); // global_prefetch_b8

    // x_t^T in B-layout (K = input index 0..2, K=3 padded)
    v2f bx;
    bx.x = x[xb + (half ? 2 : 0)];       // K=0 (lo) / K=2 (hi)
    { float x1 = x[xb + 1]; bx.y = half ? 0.0f : x1; }  // K=1 / K=3 pad

    // ---- layer 0: z0 = W_ih0 @ x^T + W_hh0 @ s0 + (b_ih0 + b_hh0) ----
    v8f c = cb0;
    c = __builtin_amdgcn_wmma_f32_16x16x4_f32(false, a_ih0,   false, bx,   (short)0, c, false, false);
    c = __builtin_amdgcn_wmma_f32_16x16x4_f32(false, a_hh0_0, false, s0c0, (short)0, c, false, false);
    c = __builtin_amdgcn_wmma_f32_16x16x4_f32(false, a_hh0_1, false, s0c1, (short)0, c, false, false);
    e0 = fast_tanh(c[0]); e1 = fast_tanh(c[1]); e2 = fast_tanh(c[2]);
    e3 = fast_tanh(c[3]); e4 = fast_tanh(c[4]);

    // D (row=vgpr, col=lane) -> B-layout for next step
    { float r2 = shfl16(e2, l16), r3 = shfl16(e3, l16);
      s0c0.x = half ? r2 : e0;
      s0c0.y = half ? r3 : e1;
      s0c1.x = half ? 0.0f : e4;
      s0c1.y = 0.0f; }

    // ---- layer 1: z1 = W_ih1 @ s0' + W_hh1 @ s1 + (b_ih1 + b_hh1) ----
    v8f d = cb1;
    d = __builtin_amdgcn_wmma_f32_16x16x4_f32(false, a_ih1_0, false, s0c0, (short)0, d, false, false);
    d = __builtin_amdgcn_wmma_f32_16x16x4_f32(false, a_ih1_1, false, s0c1, (short)0, d, false, false);
    d = __builtin_amdgcn_wmma_f32_16x16x4_f32(false, a_hh1_0, false, s1c0, (short)0, d, false, false);
    d = __builtin_amdgcn_wmma_f32_16x16x4_f32(false, a_hh1_1, false, s1c1, (short)0, d, false, false);
    f0 = fast_tanh(d[0]); f1 = fast_tanh(d[1]); f2 = fast_tanh(d[2]);
    f3 = fast_tanh(d[3]); f4 = fast_tanh(d[4]);

    { float r2 = shfl16(f2, l16), r3 = shfl16(f3, l16);
      s1c0.x = half ? r2 : f0;
      s1c0.y = half ? r3 : f1;
      s1c1.x = half ? 0.0f : f4;
      s1c1.y = 0.0f; }

    // ---- emit out[t][b][0..4] (only lanes 0-15 carry valid columns) ----
    if (!half) {
      const size_t ob = ((size_t)t * BATCH + b0 + l16) * HSZ;
      out[ob + 0] = f0; out[ob + 1] = f1; out[ob + 2] = f2;
      out[ob + 3] = f3; out[ob + 4] = f4;
    }
  }

  // ---- h_n tail: [2, B, H] appended after out[T,B,H] ----
  if (!half) {
    const size_t tail = (size_t)T_LEN * BATCH * HSZ;
    const size_t o0 = tail + (size_t)(b0 + l16) * HSZ;
    out[o0 + 0] = e0; out[o0 + 1] = e1; out[o0 + 2] = e2;
    out[o0 + 3] = e3; out[o0 + 4] = e4;
    const size_t o1 = tail + ((size_t)BATCH + b0 + l16) * HSZ;
    out[o1 + 0] = f0; out[o1 + 1] = f1; out[o1 + 2] = f2;
    out[o1 + 3] = f3; out[o1 + 4] = f4;
  }
}

extern "C" void kernel_launch(void* const* d_in, const int* in_sizes, int n_in,
                              void* d_out, int out_size, void* d_ws, size_t ws_size,
                              hipStream_t stream) {
  (void)in_sizes; (void)n_in; (void)out_size; (void)d_ws; (void)ws_size;
  const float* x     = (const float*)d_in[0];
  const float* hx    = (const float*)d_in[1];
  const float* w_ih0 = (const float*)d_in[2];
  const float* w_hh0 = (const float*)d_in[3];
  const float* b_ih0 = (const float*)d_in[4];
  const float* b_hh0 = (const float*)d_in[5];
  const float* w_ih1 = (const float*)d_in[6];
  const float* w_hh1 = (const float*)d_in[7];
  const float* b_ih1 = (const float*)d_in[8];
  const float* b_hh1 = (const float*)d_in[9];
  float* out = (float*)d_out;

  dim3 grid(BATCH / 16);  // 256 waves, one 16-batch tile each
  dim3 block(32);         // one wave32 per block
  rnn2_wmma_kernel<<<grid, block, 0, stream>>>(
      x, hx, w_ih0, w_hh0, b_ih0, b_hh0, w_ih1, w_hh1, b_ih1, b_hh1, out);
}